// CMLITargetLoss_35785667510930
// MI455X (gfx1250) — compile-verified
//
#include <hip/hip_runtime.h>
#include <hip/hip_bf16.h>

typedef float v2f __attribute__((ext_vector_type(2)));
typedef float v8f __attribute__((ext_vector_type(8)));

#define Bsz 64
#define L   197
#define D   768
#define NT  196          // tokens/patches after slicing [1:]
#define NTILE 13         // ceil(196/16)
#define LDA 772          // padded LDS row stride for text tile (floats)
#define KC  128          // K-chunk staged per async copy
#define BSTRIDE 132      // padded LDS row stride for B chunk (floats)
#define BFLT (16 * BSTRIDE)   // floats per B buffer (2112)

// 16 async b128 copies = one 16-row x 128-float chunk (8 KB) per wave.
// Lane l covers row (l>>1), byte offsets (l&1)*16 + i*32; the immediate is
// added to BOTH the global and LDS addresses, so one address pair serves all.
#define ASYNC_COPY_CHUNK(ldsaddr, gaddr)                                   \
    asm volatile(                                                          \
        "global_load_async_to_lds_b128 %0, %1, off\n\t"                    \
        "global_load_async_to_lds_b128 %0, %1, off offset:32\n\t"          \
        "global_load_async_to_lds_b128 %0, %1, off offset:64\n\t"          \
        "global_load_async_to_lds_b128 %0, %1, off offset:96\n\t"          \
        "global_load_async_to_lds_b128 %0, %1, off offset:128\n\t"         \
        "global_load_async_to_lds_b128 %0, %1, off offset:160\n\t"         \
        "global_load_async_to_lds_b128 %0, %1, off offset:192\n\t"         \
        "global_load_async_to_lds_b128 %0, %1, off offset:224\n\t"         \
        "global_load_async_to_lds_b128 %0, %1, off offset:256\n\t"         \
        "global_load_async_to_lds_b128 %0, %1, off offset:288\n\t"         \
        "global_load_async_to_lds_b128 %0, %1, off offset:320\n\t"         \
        "global_load_async_to_lds_b128 %0, %1, off offset:352\n\t"         \
        "global_load_async_to_lds_b128 %0, %1, off offset:384\n\t"         \
        "global_load_async_to_lds_b128 %0, %1, off offset:416\n\t"         \
        "global_load_async_to_lds_b128 %0, %1, off offset:448\n\t"         \
        "global_load_async_to_lds_b128 %0, %1, off offset:480"             \
        :: "v"(ldsaddr), "v"(gaddr) : "memory")

__global__ void zero_out_kernel(float* out) { out[0] = 0.0f; }

__global__ __launch_bounds__(128)
void cmli_loss_kernel(const float* __restrict__ image,
                      const float* __restrict__ text,
                      const float* __restrict__ target,
                      const int*   __restrict__ pmask,
                      float* __restrict__ out) {
    __shared__ __align__(16) float ldsA[16 * LDA];        // text tile
    __shared__ __align__(16) float ldsB[4 * 2 * BFLT];    // per-wave double buffers
    __shared__ float wbest[4][16];
    __shared__ int   widx[4][16];
    __shared__ int   t2p[16];
    __shared__ float wsum[4];

    const int tid   = threadIdx.x;
    const int lane  = tid & 31;
    const int wave  = tid >> 5;
    const int tTile = blockIdx.x;             // 0..12
    const int b     = blockIdx.y;             // 0..63

    // ---- stage text tile rows t_global = 1 + tTile*16 + row into LDS (zero-pad) ----
    for (int i = tid; i < 16 * (D / 4); i += 128) {
        int row = i / (D / 4);
        int d4  = i % (D / 4);
        int tg  = tTile * 16 + row;
        float4 v = make_float4(0.f, 0.f, 0.f, 0.f);
        if (tg < NT)
            v = *(const float4*)(text + ((size_t)b * L + 1 + tg) * D + d4 * 4);
        *(float4*)&ldsA[row * LDA + d4 * 4] = v;
    }
    __syncthreads();

    const int col = lane & 15;                // N (p) / M (t) position within tile
    const int kph = (lane >> 4) << 1;         // K sub-offset: 0 or 2

    float runBest[8];
    int   runIdx[8];
#pragma unroll
    for (int r = 0; r < 8; ++r) { runBest[r] = -__builtin_inff(); runIdx[r] = 0x7FFFFFFF; }

    const float* Alds = &ldsA[col * LDA + kph];

    // per-lane LDS destinations for the async copies (lane covers row l>>1, half l&1)
    const unsigned lB0 = (unsigned)(uintptr_t)&ldsB[wave * 2 * BFLT + (lane >> 1) * BSTRIDE]
                         + ((lane & 1) << 4);
    const unsigned lB1 = lB0 + BFLT * 4;

    // ---- each wave processes p-tiles wave, wave+4, ... ----
    for (int pt = wave; pt < NTILE; pt += 4) {
        // async-copy source row for this lane (row index l>>1 of the p-tile)
        int pcolRow = pt * 16 + (lane >> 1);
        int prowA   = (pcolRow < NT) ? (1 + pcolRow) : 1;        // clamp (masked later)
        unsigned long long gaddr =
            (unsigned long long)(uintptr_t)(image + ((size_t)b * L + prowA) * D)
            + ((lane & 1) << 4);

        v8f c0 = {0.f, 0.f, 0.f, 0.f, 0.f, 0.f, 0.f, 0.f};
        v8f c1 = {0.f, 0.f, 0.f, 0.f, 0.f, 0.f, 0.f, 0.f};

        int buf = 0;
        ASYNC_COPY_CHUNK(lB0, gaddr);          // prefetch chunk 0 into buffer 0
        gaddr += KC * 4;

        for (int chunk = 0; chunk < D / KC; ++chunk) {
            if (chunk < D / KC - 1) {
                ASYNC_COPY_CHUNK(buf ? lB0 : lB1, gaddr);        // prefetch next
                gaddr += KC * 4;
                asm volatile("s_wait_asynccnt 16" ::: "memory"); // current chunk landed
            } else {
                asm volatile("s_wait_asynccnt 0" ::: "memory");
            }

            const float* Ap = Alds + chunk * KC;
            const float* Bp = &ldsB[wave * 2 * BFLT + buf * BFLT + col * BSTRIDE + kph];
#pragma unroll 8
            for (int kk = 0; kk < KC; kk += 8) {
                v2f a0 = *(const v2f*)(Ap + kk);
                v2f b0 = *(const v2f*)(Bp + kk);
                c0 = __builtin_amdgcn_wmma_f32_16x16x4_f32(
                         false, a0, false, b0, (short)0, c0, false, false);
                v2f a1 = *(const v2f*)(Ap + kk + 4);
                v2f b1 = *(const v2f*)(Bp + kk + 4);
                c1 = __builtin_amdgcn_wmma_f32_16x16x4_f32(
                         false, a1, false, b1, (short)0, c1, false, false);
            }
            buf ^= 1;
        }

        int pcol = pt * 16 + col;                                // this lane's p column
        bool pvalid = (pcol < NT) && (pmask[b * L + 1 + pcol] == 0);
#pragma unroll
        for (int r = 0; r < 8; ++r) {
            float v = pvalid ? (c0[r] + c1[r]) : -__builtin_inff();
            if (v > runBest[r] || (v == runBest[r] && pcol < runIdx[r])) {
                runBest[r] = v; runIdx[r] = pcol;
            }
        }
    }

    // ---- argmax reduce across the 16 lanes holding one row's columns ----
#pragma unroll
    for (int r = 0; r < 8; ++r) {
        float bv = runBest[r];
        int   bi = runIdx[r];
#pragma unroll
        for (int m = 1; m < 16; m <<= 1) {
            float ov = __shfl_xor(bv, m, 32);
            int   oi = __shfl_xor(bi, m, 32);
            if (ov > bv || (ov == bv && oi < bi)) { bv = ov; bi = oi; }
        }
        if ((lane & 15) == 0) {
            int row = r + ((lane >> 4) << 3);
            wbest[wave][row] = bv;
            widx [wave][row] = bi;
        }
    }
    __syncthreads();

    // ---- combine across the 4 waves ----
    if (tid < 16) {
        float bv = wbest[0][tid];
        int   bi = widx [0][tid];
#pragma unroll
        for (int w = 1; w < 4; ++w) {
            float ov = wbest[w][tid];
            int   oi = widx [w][tid];
            if (ov > bv || (ov == bv && oi < bi)) { bv = ov; bi = oi; }
        }
        if (bi > NT - 1 || bi < 0) bi = 0;     // degenerate all-masked case
        t2p[tid] = bi;
    }
    __syncthreads();

    // ---- kd_token_loss: sum (text[b,t] - image[b,1+t2p])^2 over this tile ----
    float acc = 0.f;
    for (int row = 0; row < 16; ++row) {
        int tg = tTile * 16 + row;
        if (tg >= NT) break;
        const float* ip = image + ((size_t)b * L + 1 + t2p[row]) * D;
        for (int d = tid; d < D; d += 128) {
            float df = ldsA[row * LDA + d] - ip[d];
            acc += df * df;
        }
    }
    acc *= 1.0f / ((float)Bsz * (float)NT * (float)D);

    // ---- kd_image_cls_loss contribution (one tile-block per batch does it) ----
    if (tTile == 0) {
        const float* ip = image  + (size_t)b * L * D;
        const float* tp = target + (size_t)b * L * D;
        float a2 = 0.f;
        for (int d = tid; d < D; d += 128) {
            float df = ip[d] - tp[d];
            a2 += df * df;
        }
        acc += a2 * (1.0f / ((float)Bsz * (float)D));
    }

    // ---- block reduction + atomic accumulate ----
#pragma unroll
    for (int m = 16; m >= 1; m >>= 1) acc += __shfl_xor(acc, m, 32);
    if (lane == 0) wsum[wave] = acc;
    __syncthreads();
    if (tid == 0)
        atomicAdd(out, wsum[0] + wsum[1] + wsum[2] + wsum[3]);
}

extern "C" void kernel_launch(void* const* d_in, const int* in_sizes, int n_in,
                              void* d_out, int out_size, void* d_ws, size_t ws_size,
                              hipStream_t stream) {
    const float* image  = (const float*)d_in[0];
    const float* text   = (const float*)d_in[1];
    const float* target = (const float*)d_in[2];
    const int*   pmask  = (const int*)d_in[3];
    float* out = (float*)d_out;

    zero_out_kernel<<<1, 1, 0, stream>>>(out);
    dim3 grid(NTILE, Bsz, 1);
    cmli_loss_kernel<<<grid, 128, 0, stream>>>(image, text, target, pmask, out);
}